// Prompt_block_62139586839406
// MI455X (gfx1250) — compile-verified
//
#include <hip/hip_runtime.h>
#include <hip/hip_bf16.h>
#include <math.h>

// ---------------- problem constants ----------------
#define DIM      768
#define HIDDEN   64
#define NEXP     6
#define BB       64
#define CC       320
#define NTOK     (BB * CC)          // 20480
#define NZ       64                 // C/5
#define NPAIR    (NTOK / 32)        // 640 (32-token super-tiles)

// output layout (flat f32): x_prompted | z_prompted | loss | logits
#define XP_ELEMS ((size_t)BB * (CC - NZ) * DIM)   // 12,582,912
#define ZP_ELEMS ((size_t)BB * NZ * DIM)          //  3,145,728
#define LOSS_OFF (XP_ELEMS + ZP_ELEMS)            // 15,728,640
#define LOG_OFF  (LOSS_OFF + 1)

typedef __attribute__((ext_vector_type(2))) float v2f;
typedef __attribute__((ext_vector_type(8))) float v8f;

// ---------------- workspace layout ----------------
#define WS_GE_OFF  64
#define WS_GG_OFF  (WS_GE_OFF + (size_t)NTOK * 8)
#define WS_Y_OFF   (WS_GE_OFF + (size_t)NTOK * 16)

// ---------------- kernel 0: zero accumulators ----------------
__global__ void k_zero(float* acc) {
    if (threadIdx.x < 12) acc[threadIdx.x] = 0.0f;
}

// ---------------- kernel 1: gating ----------------
// block = 256 threads = 8 waves; one wave per token
__global__ __launch_bounds__(256) void k_gate(const float* __restrict__ x,
                                              const float* __restrict__ xi,
                                              const float* __restrict__ w_gate,
                                              float* __restrict__ logits_out,
                                              int2*  __restrict__ gate_e,
                                              float2* __restrict__ gate_g,
                                              float* __restrict__ importance,
                                              float* __restrict__ loadcnt) {
    const int wave = threadIdx.x >> 5;
    const int lane = threadIdx.x & 31;
    const int n = blockIdx.x * 8 + wave;

    const float* xr  = x  + (size_t)n * DIM;
    const float* xir = xi + (size_t)n * DIM;

    float acc[NEXP] = {0.f, 0.f, 0.f, 0.f, 0.f, 0.f};
    for (int d = lane; d < DIM; d += 32) {
        const float t = xr[d] + xir[d];
        const float* wg = w_gate + (size_t)d * NEXP;
#pragma unroll
        for (int e = 0; e < NEXP; ++e) acc[e] += t * wg[e];
    }
    // wave32 butterfly reduction
#pragma unroll
    for (int off = 16; off > 0; off >>= 1) {
#pragma unroll
        for (int e = 0; e < NEXP; ++e) acc[e] += __shfl_xor(acc[e], off, 32);
    }

    if (lane == 0) {
#pragma unroll
        for (int e = 0; e < NEXP; ++e) logits_out[(size_t)n * NEXP + e] = acc[e];
        // top-2, jax.lax.top_k tie-break: first (lowest) index wins
        int e1 = 0; float v1 = acc[0];
#pragma unroll
        for (int e = 1; e < NEXP; ++e) if (acc[e] > v1) { v1 = acc[e]; e1 = e; }
        int e2 = -1; float v2 = -1e30f;
#pragma unroll
        for (int e = 0; e < NEXP; ++e) if (e != e1 && acc[e] > v2) { v2 = acc[e]; e2 = e; }
        // softmax over (v1, v2), v1 >= v2
        const float t  = expf(v2 - v1);
        const float s  = 1.0f + t;
        gate_e[n] = make_int2(e1, e2);
        gate_g[n] = make_float2(1.0f / s, t / s);
        atomicAdd(&importance[e1], 1.0f / s);
        atomicAdd(&importance[e2], t / s);
        atomicAdd(&loadcnt[e1], 1.0f);
        atomicAdd(&loadcnt[e2], 1.0f);
    }
}

// ---------------- kernel 2: MoE expert GEMM (fp32 WMMA 16x16x4) ----------------
// block = 128 threads = 4 waves; each wave owns a 32-token super-tile (2 M-tiles
// sharing every B fragment -> half the B-load traffic per FLOP).
// A (16x4 f32): lane<16 -> row=lane, K = k0+{0,1}; lane>=16 -> row=lane-16, K = k0+{2,3}
// B (4x16 f32): VGPR v, lane -> B[k0 + 2*(lane>>4) + v, lane&15]
// C/D (16x16):  VGPR i, lane -> D[i + 8*(lane>>4), lane&15]
__global__ __launch_bounds__(128) void k_expert(const float* __restrict__ x,
                                                const float* __restrict__ xi,
                                                const float* __restrict__ expert_w,
                                                const float* __restrict__ expert_b,
                                                const int2*  __restrict__ gate_e,
                                                const float2* __restrict__ gate_g,
                                                float* __restrict__ y) {
    const int wave  = threadIdx.x >> 5;
    const int lane  = threadIdx.x & 31;
    const int row   = lane & 15;
    const int khalf = lane >> 4;
    const int pair  = blockIdx.x * 4 + wave;
    const int n0    = pair * 32;

    // dense per-token gate table: straight-line ds_load at scale time
    __shared__ float s_gate[4][32][NEXP];
    {
        const int n = n0 + lane;
#pragma unroll
        for (int e = 0; e < NEXP; ++e) s_gate[wave][lane][e] = 0.0f;
        const int2   ee = gate_e[n];
        const float2 gg = gate_g[n];
        s_gate[wave][lane][ee.x] = gg.x;
        s_gate[wave][lane][ee.y] = gg.y;
    }
    __syncthreads();

    const float* xr0  = x  + (size_t)(n0 + row) * DIM;
    const float* xir0 = xi + (size_t)(n0 + row) * DIM;
    const float* xr1  = xr0  + (size_t)16 * DIM;
    const float* xir1 = xir0 + (size_t)16 * DIM;

    v8f yacc[2][4] = {};   // 32x64 output super-tile

    for (int e = 0; e < NEXP; ++e) {
        const float* W = expert_w + (size_t)e * DIM * HIDDEN;
        v8f acc[2][4] = {};
        for (int k0 = 0; k0 < DIM; k0 += 4) {
            const int kk = k0 + 2 * khalf;
            if ((k0 & 63) == 0 && k0 + 128 < DIM) {
                __builtin_prefetch(xr0 + k0 + 128, 0, 3);   // global_prefetch_b8
                __builtin_prefetch(xr1 + k0 + 128, 0, 3);
            }
            v2f a0, a1;
            a0.x = xr0[kk]     + xir0[kk];
            a0.y = xr0[kk + 1] + xir0[kk + 1];
            a1.x = xr1[kk]     + xir1[kk];
            a1.y = xr1[kk + 1] + xir1[kk + 1];
#pragma unroll
            for (int t = 0; t < 4; ++t) {
                const int h = t * 16 + row;
                v2f b;
                b.x = W[(size_t)kk * HIDDEN + h];
                b.y = W[(size_t)(kk + 1) * HIDDEN + h];
                acc[0][t] = __builtin_amdgcn_wmma_f32_16x16x4_f32(
                    false, a0, false, b, (short)0, acc[0][t], false, false);
                acc[1][t] = __builtin_amdgcn_wmma_f32_16x16x4_f32(
                    false, a1, false, b, (short)0, acc[1][t], false, false);
            }
        }
        // y += gate[n,e] * (acc + expert_b[e,h]); gate via dense LDS table
        float gv[2][8];
#pragma unroll
        for (int m = 0; m < 2; ++m)
#pragma unroll
            for (int i = 0; i < 8; ++i)
                gv[m][i] = s_gate[wave][m * 16 + i + 8 * khalf][e];
#pragma unroll
        for (int m = 0; m < 2; ++m)
#pragma unroll
            for (int t = 0; t < 4; ++t) {
                const float be = expert_b[e * HIDDEN + t * 16 + row];
#pragma unroll
                for (int i = 0; i < 8; ++i)
                    yacc[m][t][i] += gv[m][i] * (acc[m][t][i] + be);
            }
    }
#pragma unroll
    for (int m = 0; m < 2; ++m)
#pragma unroll
        for (int t = 0; t < 4; ++t)
#pragma unroll
            for (int i = 0; i < 8; ++i)
                y[(size_t)(n0 + m * 16 + i + 8 * khalf) * HIDDEN + t * 16 + row] =
                    yacc[m][t][i];
}

// ---------------- kernel 3: FFN out = y @ ffn1_w^T + b (fp32 WMMA) ----------------
// grid (NPAIR, 3), block 128 = 4 waves; wave -> 64-col d-chunk of a 32-token tile
__global__ __launch_bounds__(128) void k_ffn(const float* __restrict__ y,
                                             const float* __restrict__ ffn1_w,
                                             const float* __restrict__ ffn1_b,
                                             float* __restrict__ out_x,
                                             float* __restrict__ out_z) {
    const int wave  = threadIdx.x >> 5;
    const int lane  = threadIdx.x & 31;
    const int row   = lane & 15;
    const int khalf = lane >> 4;
    const int n0    = blockIdx.x * 32;
    const int d0    = (blockIdx.y * 4 + wave) * 64;

    const float* yr0 = y + (size_t)(n0 + row) * HIDDEN;
    const float* yr1 = yr0 + (size_t)16 * HIDDEN;

    v8f acc[2][4] = {};
    for (int k0 = 0; k0 < HIDDEN; k0 += 4) {
        const int kk = k0 + 2 * khalf;
        v2f a0, a1;
        a0.x = yr0[kk];
        a0.y = yr0[kk + 1];
        a1.x = yr1[kk];
        a1.y = yr1[kk + 1];
#pragma unroll
        for (int t = 0; t < 4; ++t) {
            const int d = d0 + t * 16 + row;
            v2f b;                               // B[k, d] = ffn1_w[d, k]
            b.x = ffn1_w[(size_t)d * HIDDEN + kk];
            b.y = ffn1_w[(size_t)d * HIDDEN + kk + 1];
            acc[0][t] = __builtin_amdgcn_wmma_f32_16x16x4_f32(
                false, a0, false, b, (short)0, acc[0][t], false, false);
            acc[1][t] = __builtin_amdgcn_wmma_f32_16x16x4_f32(
                false, a1, false, b, (short)0, acc[1][t], false, false);
        }
    }
#pragma unroll
    for (int t = 0; t < 4; ++t) {
        const int d = d0 + t * 16 + row;
        const float bias = ffn1_b[d];
#pragma unroll
        for (int m = 0; m < 2; ++m)
#pragma unroll
            for (int i = 0; i < 8; ++i) {
                const int n = n0 + m * 16 + i + 8 * khalf;
                const int b = n / CC;
                const int c = n % CC;
                const float v = acc[m][t][i] + bias;
                if (c < NZ)
                    out_z[((size_t)b * NZ + c) * DIM + d] = v;
                else
                    out_x[((size_t)b * (CC - NZ) + (c - NZ)) * DIM + d] = v;
            }
    }
}

// ---------------- kernel 4: loss = cv2(importance) + cv2(load) ----------------
__global__ void k_loss(const float* __restrict__ imp, const float* __restrict__ lod,
                       float* __restrict__ loss_out) {
    float mi = 0.f, ml = 0.f;
    for (int e = 0; e < NEXP; ++e) { mi += imp[e]; ml += lod[e]; }
    mi /= NEXP; ml /= NEXP;
    float vi = 0.f, vl = 0.f;
    for (int e = 0; e < NEXP; ++e) {
        const float di = imp[e] - mi, dl = lod[e] - ml;
        vi += di * di; vl += dl * dl;
    }
    vi /= NEXP; vl /= NEXP;
    *loss_out = vi / (mi * mi + 1e-10f) + vl / (ml * ml + 1e-10f);
}

extern "C" void kernel_launch(void* const* d_in, const int* in_sizes, int n_in,
                              void* d_out, int out_size, void* d_ws, size_t ws_size,
                              hipStream_t stream) {
    const float* x        = (const float*)d_in[0];
    const float* xi       = (const float*)d_in[1];
    const float* w_gate   = (const float*)d_in[2];
    const float* expert_w = (const float*)d_in[3];
    const float* expert_b = (const float*)d_in[4];
    const float* ffn1_w   = (const float*)d_in[5];
    const float* ffn1_b   = (const float*)d_in[6];

    float* out = (float*)d_out;
    float* out_x   = out;
    float* out_z   = out + XP_ELEMS;
    float* out_ls  = out + LOSS_OFF;
    float* out_log = out + LOG_OFF;

    float*  imp = (float*)d_ws;
    float*  lod = imp + 6;
    int2*   ge  = (int2*)  ((char*)d_ws + WS_GE_OFF);
    float2* gg  = (float2*)((char*)d_ws + WS_GG_OFF);
    float*  y   = (float*) ((char*)d_ws + WS_Y_OFF);

    k_zero<<<1, 32, 0, stream>>>(imp);
    k_gate<<<NTOK / 8, 256, 0, stream>>>(x, xi, w_gate, out_log, ge, gg, imp, lod);
    k_expert<<<NPAIR / 4, 128, 0, stream>>>(x, xi, expert_w, expert_b, ge, gg, y);
    k_ffn<<<dim3(NPAIR, 3), 128, 0, stream>>>(y, ffn1_w, ffn1_b, out_x, out_z);
    k_loss<<<1, 1, 0, stream>>>(imp, lod, out_ls);
}